// EPALite_58463094833350
// MI455X (gfx1250) — compile-verified
//
#include <hip/hip_runtime.h>
#include <math.h>

// ---------------------------------------------------------------------------
// EPA-Lite attention for MI455X (gfx1250), fp32 throughout using
// V_WMMA_F32_16X16X4_F32 (wave32, 16x16 tiles, K-step 4).
// B=4, N=8192, C=512, H=8, d=64, P=256.
// ---------------------------------------------------------------------------

typedef __attribute__((ext_vector_type(2))) float v2f;
typedef __attribute__((ext_vector_type(8))) float v8f;

__device__ __forceinline__ v8f wmma_f32(v2f a, v2f b, v8f c) {
  // 8-arg pattern: (neg_a, A, neg_b, B, c_mod, C, reuse_a, reuse_b)
  return __builtin_amdgcn_wmma_f32_16x16x4_f32(false, a, false, b, (short)0, c,
                                               false, false);
}

// ===========================================================================
// K1: qkvv = x @ w_qkvv   [32768 x 512] @ [512 x 2048]
// Epilogue scatters into q/k/vca/vsa with [B,H,d,N] layout.
// ===========================================================================
__global__ __launch_bounds__(256) void k_qkvv(const float* __restrict__ X,
                                              const float* __restrict__ W,
                                              float* __restrict__ q,
                                              float* __restrict__ kk,
                                              float* __restrict__ vca,
                                              float* __restrict__ vsa) {
  __shared__ float As[64][34];  // [m][k], pad 2 -> conflict-free frag reads
  __shared__ float Bs[64][34];  // transposed [n][k]
  const int tid = threadIdx.x, lane = tid & 31, wv = tid >> 5;
  const int tm = blockIdx.x * 64, tn = blockIdx.y * 64;
  const int wm = (wv & 3) * 16, wn = (wv >> 2) * 32;
  v8f acc0 = {}; v8f acc1 = {};
  for (int k0 = 0; k0 < 512; k0 += 32) {
    for (int i = tid; i < 2048; i += 256) {
      int r = i >> 5, c = i & 31;
      As[r][c] = X[(size_t)(tm + r) * 512 + k0 + c];
    }
    for (int i = tid; i < 2048; i += 256) {
      int r = i >> 6, c = i & 63;
      Bs[c][r] = W[(size_t)(k0 + r) * 2048 + tn + c];
    }
    __syncthreads();
    const int mr = wm + (lane & 15), nc = wn + (lane & 15);
#pragma unroll
    for (int kx = 0; kx < 32; kx += 4) {
      int ko = kx + ((lane >> 4) << 1);           // lanes>=16 take K=2,3
      v2f a  = *(const v2f*)&As[mr][ko];
      v2f b0 = *(const v2f*)&Bs[nc][ko];
      v2f b1 = *(const v2f*)&Bs[nc + 16][ko];
      acc0 = wmma_f32(a, b0, acc0);
      acc1 = wmma_f32(a, b1, acc1);
    }
    __syncthreads();
  }
  // scatter: col c -> (s = c>>9, h = (c>>6)&7, e = c&63); dest[s][b,h,e,n]
  float* dst[4] = {q, kk, vca, vsa};
  const int trow = tm + wm + ((lane >> 4) << 3);  // 8 consecutive tokens
  const int b = trow >> 13, n0 = trow & 8191;
#pragma unroll
  for (int t2 = 0; t2 < 2; ++t2) {
    int c = tn + wn + t2 * 16 + (lane & 15);
    int s = c >> 9, h = (c >> 6) & 7, e = c & 63;
    float* dp = dst[s] + ((size_t)((b * 8 + h) * 64 + e)) * 8192 + n0;
    v8f acc = t2 ? acc1 : acc0;
    float4 lo; lo.x = acc[0]; lo.y = acc[1]; lo.z = acc[2]; lo.w = acc[3];
    float4 hi; hi.x = acc[4]; hi.y = acc[5]; hi.z = acc[6]; hi.w = acc[7];
    *(float4*)(dp) = lo;
    *(float4*)(dp + 4) = hi;
  }
}

// ===========================================================================
// K2: per row (b,h,e): 1/||q||, 1/||k||, pooled kp (normalized), pooled vp.
// One block per row; each wave's 32 lanes == one pooling window.
// ===========================================================================
__global__ __launch_bounds__(256) void k_normpool(const float* __restrict__ q,
                                                  const float* __restrict__ k,
                                                  const float* __restrict__ vsa,
                                                  float* __restrict__ rq,
                                                  float* __restrict__ rk,
                                                  float* __restrict__ kp,
                                                  float* __restrict__ vp) {
  __shared__ float poolk[256], poolv[256];
  __shared__ float redq[8], redk[8], bc[1];
  const int row = blockIdx.x;  // (b*8+h)*64+e
  const float* qr = q + (size_t)row * 8192;
  const float* kr = k + (size_t)row * 8192;
  const float* vr = vsa + (size_t)row * 8192;
  const int t = threadIdx.x, lane = t & 31, wv = t >> 5;
  float sq = 0.f, sk = 0.f;
  for (int i = 0; i < 32; ++i) {
    int idx = i * 256 + t;                      // coalesced
    float qa = qr[idx]; sq += qa * qa;
    float ka = kr[idx]; sk += ka * ka;
    float va = vr[idx];
    float ks2 = ka, vs2 = va;                   // window = i*8 + wv
#pragma unroll
    for (int off = 16; off; off >>= 1) {
      ks2 += __shfl_xor(ks2, off, 32);
      vs2 += __shfl_xor(vs2, off, 32);
    }
    if (lane == 0) {
      poolk[i * 8 + wv] = ks2 * (1.f / 32.f);
      poolv[i * 8 + wv] = vs2 * (1.f / 32.f);
    }
  }
#pragma unroll
  for (int off = 16; off; off >>= 1) {
    sq += __shfl_xor(sq, off, 32);
    sk += __shfl_xor(sk, off, 32);
  }
  if (lane == 0) { redq[wv] = sq; redk[wv] = sk; }
  __syncthreads();
  if (t == 0) {
    float a = 0.f, c = 0.f;
    for (int w = 0; w < 8; ++w) { a += redq[w]; c += redk[w]; }
    float irq = 1.f / fmaxf(sqrtf(a), 1e-12f);
    float irk = 1.f / fmaxf(sqrtf(c), 1e-12f);
    rq[row] = irq; rk[row] = irk; bc[0] = irk;
  }
  __syncthreads();
  float irk = bc[0];
  kp[(size_t)row * 256 + t] = poolk[t] * irk;   // pool(k/||k||)=pool(k)/||k||
  vp[(size_t)row * 256 + t] = poolv[t];
}

// ===========================================================================
// K3: channel attention, one block per (b,h).
//   S = (q kT) * rq[e]*rk[f]*tau; softmax rows; x_ca = S @ v_ca.
//   x_ca stored as [B,N,C] (channel-contiguous per lane -> b128 stores).
// ===========================================================================
__global__ __launch_bounds__(256) void k_chan(const float* __restrict__ q,
                                              const float* __restrict__ k,
                                              const float* __restrict__ v,
                                              const float* __restrict__ rq,
                                              const float* __restrict__ rk,
                                              const float* __restrict__ temp,
                                              float* __restrict__ xca) {
  __shared__ float Qs[64][34];
  __shared__ float Ks[64][34];
  __shared__ float Sm[64][66];
  __shared__ float rqs[64], rks[64];
  const int bh = blockIdx.x, b = bh >> 3, h = bh & 7;
  const float* qb = q + (size_t)bh * 64 * 8192;
  const float* kb = k + (size_t)bh * 64 * 8192;
  const float* vb = v + (size_t)bh * 64 * 8192;
  const int tid = threadIdx.x, lane = tid & 31, wv = tid >> 5;
  const int wm = (wv & 3) * 16, wn = (wv >> 2) * 32;
  if (tid < 64) { rqs[tid] = rq[bh * 64 + tid]; rks[tid] = rk[bh * 64 + tid]; }
  v8f acc0 = {}; v8f acc1 = {};
  for (int k0 = 0; k0 < 8192; k0 += 32) {       // S = q @ kT, K = N
    for (int i = tid; i < 2048; i += 256) {
      int r = i >> 5, c = i & 31;
      Qs[r][c] = qb[(size_t)r * 8192 + k0 + c];
      Ks[r][c] = kb[(size_t)r * 8192 + k0 + c];
    }
    __syncthreads();
    const int mr = wm + (lane & 15), nc = wn + (lane & 15);
#pragma unroll
    for (int kx = 0; kx < 32; kx += 4) {
      int ko = kx + ((lane >> 4) << 1);
      v2f a  = *(const v2f*)&Qs[mr][ko];
      v2f b0 = *(const v2f*)&Ks[nc][ko];
      v2f b1 = *(const v2f*)&Ks[nc + 16][ko];
      acc0 = wmma_f32(a, b0, acc0);
      acc1 = wmma_f32(a, b1, acc1);
    }
    __syncthreads();
  }
  const float tscale = temp[h];
  {
    int ebase = wm + ((lane >> 4) << 3);
#pragma unroll
    for (int t2 = 0; t2 < 2; ++t2) {
      int f = wn + t2 * 16 + (lane & 15);
      v8f acc = t2 ? acc1 : acc0;
#pragma unroll
      for (int r = 0; r < 8; ++r) {
        int e = ebase + r;
        Sm[e][f] = acc[r] * rqs[e] * rks[f] * tscale;
      }
    }
  }
  __syncthreads();
  if (tid < 64) {                               // softmax over f (64 wide)
    float m = -3.4e38f;
    for (int j = 0; j < 64; ++j) m = fmaxf(m, Sm[tid][j]);
    float s = 0.f;
    for (int j = 0; j < 64; ++j) { float ex = __expf(Sm[tid][j] - m); Sm[tid][j] = ex; s += ex; }
    float inv = 1.f / s;
    for (int j = 0; j < 64; ++j) Sm[tid][j] *= inv;
  }
  __syncthreads();
  // x_ca = Sm @ v_ca : [64 x 8192], B fragments coalesced from global
  const int mr = wm + (lane & 15);
  const int ebase2 = wm + ((lane >> 4) << 3);
  for (int n0 = 0; n0 < 8192; n0 += 64) {
    v8f c0 = {}; v8f c1 = {};
    int col0 = n0 + wn + (lane & 15);
#pragma unroll
    for (int kx = 0; kx < 64; kx += 4) {
      int ko = kx + ((lane >> 4) << 1);
      v2f a = *(const v2f*)&Sm[mr][ko];
      v2f b0, b1;
      b0[0] = vb[(size_t)ko * 8192 + col0];
      b0[1] = vb[(size_t)(ko + 1) * 8192 + col0];
      b1[0] = vb[(size_t)ko * 8192 + col0 + 16];
      b1[1] = vb[(size_t)(ko + 1) * 8192 + col0 + 16];
      c0 = wmma_f32(a, b0, c0);
      c1 = wmma_f32(a, b1, c1);
    }
#pragma unroll
    for (int t2 = 0; t2 < 2; ++t2) {
      int tok = n0 + wn + t2 * 16 + (lane & 15);
      v8f acc = t2 ? c1 : c0;
      float* dp = xca + ((size_t)b * 8192 + tok) * 512 + h * 64 + ebase2;
      float4 lo; lo.x = acc[0]; lo.y = acc[1]; lo.z = acc[2]; lo.w = acc[3];
      float4 hi; hi.x = acc[4]; hi.y = acc[5]; hi.z = acc[6]; hi.w = acc[7];
      *(float4*)dp = lo;
      *(float4*)(dp + 4) = hi;
    }
  }
}

// ===========================================================================
// K4: spatial attention, one block per (b,h, 32-token tile).
//   L[32,256] = (rq*q)^T @ kp ; softmax over p ; x_sa = L @ vp^T -> [32,64]
// ===========================================================================
__global__ __launch_bounds__(256) void k_spat(const float* __restrict__ q,
                                              const float* __restrict__ rq,
                                              const float* __restrict__ kp,
                                              const float* __restrict__ vp,
                                              const float* __restrict__ temp2,
                                              float* __restrict__ xsa) {
  __shared__ float L[32][258];
  __shared__ float rqs[64];
  const int blk = blockIdx.x;
  const int bh = blk >> 8, tb = blk & 255;      // 256 token tiles of 32
  const int b = bh >> 3, h = bh & 7;
  const int n0 = tb * 32;
  const float* qb = q + (size_t)bh * 64 * 8192;
  const float* kpb = kp + (size_t)bh * 64 * 256;
  const float* vpb = vp + (size_t)bh * 64 * 256;
  const int tid = threadIdx.x, lane = tid & 31, wv = tid >> 5;
  if (tid < 64) rqs[tid] = rq[bh * 64 + tid];
  __syncthreads();
  // logits: M=32 tokens, N=256 pools, K=64 channels
  const int wm = (wv & 1) * 16;
  const int wc = (wv >> 1) * 64;
  const int mtok = wm + (lane & 15);
  v8f acc[4] = {};
#pragma unroll
  for (int kx = 0; kx < 64; kx += 4) {
    int ko = kx + ((lane >> 4) << 1);
    v2f a;                                      // coalesced across lanes
    a[0] = qb[(size_t)ko * 8192 + n0 + mtok] * rqs[ko];
    a[1] = qb[(size_t)(ko + 1) * 8192 + n0 + mtok] * rqs[ko + 1];
#pragma unroll
    for (int j = 0; j < 4; ++j) {
      int pc = wc + j * 16 + (lane & 15);
      v2f bf;
      bf[0] = kpb[(size_t)ko * 256 + pc];
      bf[1] = kpb[(size_t)(ko + 1) * 256 + pc];
      acc[j] = wmma_f32(a, bf, acc[j]);
    }
  }
  const float t2s = temp2[h];
  {
    int rbase = wm + ((lane >> 4) << 3);
#pragma unroll
    for (int j = 0; j < 4; ++j) {
      int pc = wc + j * 16 + (lane & 15);
#pragma unroll
      for (int r = 0; r < 8; ++r) L[rbase + r][pc] = acc[j][r] * t2s;
    }
  }
  __syncthreads();
  if (tid < 32) {                               // softmax over 256 pools
    float m = -3.4e38f;
    for (int j = 0; j < 256; ++j) m = fmaxf(m, L[tid][j]);
    float s = 0.f;
    for (int j = 0; j < 256; ++j) { float ex = __expf(L[tid][j] - m); L[tid][j] = ex; s += ex; }
    float inv = 1.f / s;
    for (int j = 0; j < 256; ++j) L[tid][j] *= inv;
  }
  __syncthreads();
  // x_sa = L @ vp^T : M=32 tokens, N=64 channels, K=256 pools
  const int we = (wv >> 1) * 16;
  const int ecol = we + (lane & 15);
  v8f o = {};
#pragma unroll 4
  for (int kx = 0; kx < 256; kx += 4) {
    int ko = kx + ((lane >> 4) << 1);
    v2f a = *(const v2f*)&L[wm + (lane & 15)][ko];
    v2f bf = *(const v2f*)(vpb + (size_t)ecol * 256 + ko);  // per-lane v2f
    o = wmma_f32(a, bf, o);
  }
  {
    int rbase = wm + ((lane >> 4) << 3);
#pragma unroll
    for (int r = 0; r < 8; ++r) {
      int tok = n0 + rbase + r;
      xsa[((size_t)b * 8192 + tok) * 512 + h * 64 + ecol] = o[r];
    }
  }
}

// ===========================================================================
// K5: output projection  Y[:, coff:coff+256] = A @ W + bias
//   A: [32768 x 512], W: [512 x 256], Y: [32768 x 512]
// ===========================================================================
__global__ __launch_bounds__(256) void k_proj(const float* __restrict__ A,
                                              const float* __restrict__ W,
                                              const float* __restrict__ bias,
                                              float* __restrict__ Y,
                                              int coff) {
  __shared__ float As[64][34];
  __shared__ float Bs[64][34];
  const int tid = threadIdx.x, lane = tid & 31, wv = tid >> 5;
  const int tm = blockIdx.x * 64, tn = blockIdx.y * 64;
  const int wm = (wv & 3) * 16, wn = (wv >> 2) * 32;
  v8f acc0 = {}; v8f acc1 = {};
  for (int k0 = 0; k0 < 512; k0 += 32) {
    for (int i = tid; i < 2048; i += 256) {
      int r = i >> 5, c = i & 31;
      As[r][c] = A[(size_t)(tm + r) * 512 + k0 + c];
    }
    for (int i = tid; i < 2048; i += 256) {
      int r = i >> 6, c = i & 63;
      Bs[c][r] = W[(size_t)(k0 + r) * 256 + tn + c];
    }
    __syncthreads();
    const int mr = wm + (lane & 15), nc = wn + (lane & 15);
#pragma unroll
    for (int kx = 0; kx < 32; kx += 4) {
      int ko = kx + ((lane >> 4) << 1);
      v2f a  = *(const v2f*)&As[mr][ko];
      v2f b0 = *(const v2f*)&Bs[nc][ko];
      v2f b1 = *(const v2f*)&Bs[nc + 16][ko];
      acc0 = wmma_f32(a, b0, acc0);
      acc1 = wmma_f32(a, b1, acc1);
    }
    __syncthreads();
  }
  const int rbase = tm + wm + ((lane >> 4) << 3);
#pragma unroll
  for (int t2 = 0; t2 < 2; ++t2) {
    int c = tn + wn + t2 * 16 + (lane & 15);
    float bv = bias[c];
    v8f acc = t2 ? acc1 : acc0;
#pragma unroll
    for (int r = 0; r < 8; ++r)
      Y[(size_t)(rbase + r) * 512 + coff + c] = acc[r] + bv;
  }
}

// ===========================================================================
extern "C" void kernel_launch(void* const* d_in, const int* in_sizes, int n_in,
                              void* d_out, int out_size, void* d_ws,
                              size_t ws_size, hipStream_t stream) {
  const float* x      = (const float*)d_in[0];
  const float* w_qkvv = (const float*)d_in[1];
  const float* w_out  = (const float*)d_in[2];
  const float* b_out  = (const float*)d_in[3];
  const float* w_out2 = (const float*)d_in[4];
  const float* b_out2 = (const float*)d_in[5];
  const float* temp   = (const float*)d_in[6];
  const float* temp2  = (const float*)d_in[7];
  float* y  = (float*)d_out;
  float* ws = (float*)d_ws;

  const size_t S = (size_t)4 * 8 * 64 * 8192;   // per-tensor [B,H,d,N]
  float* q   = ws;
  float* k   = ws + S;
  float* vca = ws + 2 * S;
  float* vsa = ws + 3 * S;
  float* xca = ws + 4 * S;                      // [B,N,C]
  float* xsa = ws + 5 * S;                      // [B,N,C]
  float* rq  = ws + 6 * S;                      // 2048
  float* rk  = rq + 2048;
  float* kp  = rk + 2048;                       // [B,H,d,P] = 524288
  float* vp  = kp + 524288;

  k_qkvv<<<dim3(32768 / 64, 2048 / 64), 256, 0, stream>>>(x, w_qkvv, q, k, vca, vsa);
  k_normpool<<<2048, 256, 0, stream>>>(q, k, vsa, rq, rk, kp, vp);
  k_chan<<<32, 256, 0, stream>>>(q, k, vca, rq, rk, temp, xca);
  k_spat<<<32 * 256, 256, 0, stream>>>(q, rq, kp, vp, temp2, xsa);
  k_proj<<<dim3(512, 4), 256, 0, stream>>>(xsa, w_out, b_out, y, 0);
  k_proj<<<dim3(512, 4), 256, 0, stream>>>(xca, w_out2, b_out2, y, 256);
}